// AdjStackAttentionWeights_12799002542745
// MI455X (gfx1250) — compile-verified
//
#include <hip/hip_runtime.h>
#include <stdint.h>

typedef __attribute__((ext_vector_type(2))) float v2f;
typedef __attribute__((ext_vector_type(8))) float v8f;

// Problem constants (from reference): B=16, S=16, N=512, H=8
constexpr int Bc = 16;
constexpr int Sc = 16;
constexpr int Nc = 512;
constexpr int Hc = 8;
constexpr int NN = Nc * Nc;                       // 262144 pixels per (b, s) plane
constexpr int TILES_PER_WAVE = 8;                 // 8 tiles x 16 pixels = 128 px / wave
constexpr int PIX_PER_WAVE = 16 * TILES_PER_WAVE; // 128
constexpr int WAVES_PER_BATCH = NN / PIX_PER_WAVE;// 2048
constexpr int BLOCK = 256;                        // 8 wave32s per block
constexpr int GRID = (Bc * WAVES_PER_BATCH * 32) / BLOCK; // 4096 blocks

__global__ __launch_bounds__(BLOCK)
void adjstack_wmma_f32_kernel(const float* __restrict__ stacks,   // [B,S,N,N]
                              const uint8_t* __restrict__ mask,   // [B,N,N] (bool)
                              const float* __restrict__ W,        // [H,S]
                              const float* __restrict__ bias,     // [H]
                              float* __restrict__ out)            // [B,H,N,N]
{
    const int lane = threadIdx.x & 31;
    const int lo   = lane & 15;       // column / row-in-half index
    const int hi   = lane >> 4;       // which half of the wave (K split)
    const int wave = blockIdx.x * (BLOCK >> 5) + (threadIdx.x >> 5);

    const int bb     = wave / WAVES_PER_BATCH;
    const int pbase0 = (wave % WAVES_PER_BATCH) * PIX_PER_WAVE;

    // ---- A fragment: W padded to 16x16 (rows >= H are zero), four K=4 chunks.
    // ISA f32 16x4 A layout: lane m (0-15) holds K=0,1 in vgpr0,1; lane 16+m holds K=2,3.
    v2f a[4];
    const int m = lo;
    #pragma unroll
    for (int c = 0; c < 4; ++c) {
        const int k = 4 * c + 2 * hi;
        a[c].x = (m < Hc) ? W[m * Sc + k]     : 0.0f;
        a[c].y = (m < Hc) ? W[m * Sc + k + 1] : 0.0f;
    }

    // Bias per output row r (wave-uniform loads; tiny, cached)
    float brow[Hc];
    #pragma unroll
    for (int r = 0; r < Hc; ++r) brow[r] = bias[r];

    const float*   sbase = stacks + (size_t)bb * Sc * NN;
    const uint8_t* mbase = mask   + (size_t)bb * NN;
    float*         obase = out    + (size_t)bb * Hc * NN;

    #pragma unroll
    for (int t = 0; t < TILES_PER_WAVE; ++t) {
        const int p = pbase0 + 16 * t + lo;   // pixel column for this lane

        // ---- B fragment per chunk: 4x16 (KxN) f32; lane n holds K=0,1 (hi=0)
        // or K=2,3 (hi=1) of the chunk. Streaming data -> non-temporal loads.
        v8f acc = {};
        #pragma unroll
        for (int c = 0; c < 4; ++c) {
            const int s = 4 * c + 2 * hi;
            v2f bfrag;
            bfrag.x = __builtin_nontemporal_load(sbase + (size_t)s       * NN + p);
            bfrag.y = __builtin_nontemporal_load(sbase + (size_t)(s + 1) * NN + p);
            // D = A(chunk c) x B(chunk c) + acc   (fp32-exact, matrix pipe)
            acc = __builtin_amdgcn_wmma_f32_16x16x4_f32(
                      /*neg_a=*/false, a[c],
                      /*neg_b=*/false, bfrag,
                      /*c_mod=*/(short)0, acc,
                      /*reuse_a=*/false, /*reuse_b=*/false);
        }

        const bool masked = (mbase[pbase0 + 16 * t + lo] != 0);

        // C/D layout: rows 0..7 live in vgprs 0..7 of lanes 0..15 (rows 8..15
        // are zero because A rows >= H were zeroed). 8 contiguous 64B NT stores.
        if (hi == 0) {
            #pragma unroll
            for (int r = 0; r < Hc; ++r) {
                const float v = masked ? 0.0f : (acc[r] + brow[r]);
                __builtin_nontemporal_store(v, obase + (size_t)r * NN + p);
            }
        }
    }
}

extern "C" void kernel_launch(void* const* d_in, const int* in_sizes, int n_in,
                              void* d_out, int out_size, void* d_ws, size_t ws_size,
                              hipStream_t stream) {
    (void)in_sizes; (void)n_in; (void)out_size; (void)d_ws; (void)ws_size;
    const float*   stacks = (const float*)d_in[0];   // [16,16,512,512] f32
    const uint8_t* mask   = (const uint8_t*)d_in[1]; // [16,512,512] bool (1B)
    const float*   W      = (const float*)d_in[2];   // [8,16] f32
    const float*   bias   = (const float*)d_in[3];   // [8] f32
    float*         out    = (float*)d_out;           // [16,8,512,512] f32

    hipLaunchKernelGGL(adjstack_wmma_f32_kernel,
                       dim3(GRID), dim3(BLOCK), 0, stream,
                       stacks, mask, W, bias, out);
}